// CorticalSheet_80633716015350
// MI455X (gfx1250) — compile-verified
//
#include <hip/hip_runtime.h>
#include <stdint.h>

// Fixed-degree SpMM on MI455X (gfx1250):
//   out[n,b] = sum_k values[n,k] * x[indices[n,k], b] + bias[n]
// N = 1e6, K = 32, B = 8.  Bandwidth-bound (~324 MB HBM => ~14us floor at
// 23.3 TB/s); x (32 MB) fits in the 192 MB L2, so gathers are L2 traffic.
// WMMA does not apply (per-row distinct gathered B operand), so the CDNA5
// feature used is the async global->LDS pipeline (GLOBAL_LOAD_ASYNC_TO_LDS_B128
// + s_wait_asynccnt) to stream values/indices with double buffering.
// Cache policy: the 256 MB values/indices stream is loaded NT so it does not
// evict the x working set from L2; out rows are stored non-temporally.

#define TILE 128                 // neurons per tile == threads per block
#define C16_PER_TILE (TILE * 8)  // float4/int4 slots per tile (K=32 -> 8 per neuron)

typedef float f32x4 __attribute__((ext_vector_type(4)));

// XOR swizzle: spreads the per-neuron 128B rows across LDS banks so the
// ds_load_b128 read pattern (lane t, chunk j) is 2-pass conflict-free.
__device__ __forceinline__ int swz(int s) { return s ^ ((s >> 4) & 7); }

// Async global->LDS copy of 16 bytes per lane. GV mode: 64-bit global address
// in a VGPR pair, 32-bit LDS byte address in a VGPR (low 32 bits of the
// generic pointer to a __shared__ object are the LDS offset). NT hint keeps
// the once-read stream from displacing x in L2.
__device__ __forceinline__ void async_b128_nt(const void* g, void* l) {
  unsigned lofs = (unsigned)(uintptr_t)l;
  unsigned long long ga = (unsigned long long)(uintptr_t)g;
  asm volatile("global_load_async_to_lds_b128 %0, %1, off th:TH_LOAD_NT"
               :: "v"(lofs), "v"(ga) : "memory");
}

template <int CNT>
__device__ __forceinline__ void wait_async() {
#if __has_builtin(__builtin_amdgcn_s_wait_asynccnt)
  __builtin_amdgcn_s_wait_asynccnt(CNT);
#else
  asm volatile("s_wait_asynccnt %0" :: "i"(CNT));
#endif
}

// NOTE: parameter must not be named 'w' (would be substituted into lo.w/hi.w).
#define GATHER_FMA(wt, idv)                                        \
  {                                                                \
    const float4* xr = gx + (size_t)(unsigned)(idv) * 2;           \
    float4 lo = xr[0];                                             \
    float4 hi = xr[1];                                             \
    a0 = fmaf((wt), lo.x, a0); a1 = fmaf((wt), lo.y, a1);          \
    a2 = fmaf((wt), lo.z, a2); a3 = fmaf((wt), lo.w, a3);          \
    a4 = fmaf((wt), hi.x, a4); a5 = fmaf((wt), hi.y, a5);          \
    a6 = fmaf((wt), hi.z, a6); a7 = fmaf((wt), hi.w, a7);          \
  }

__global__ __launch_bounds__(TILE) void CorticalSheet_spmm_kernel(
    const float* __restrict__ x, const float* __restrict__ values,
    const float* __restrict__ bias, const int* __restrict__ indices,
    float* __restrict__ out, int N, int nTiles) {
  // Double-buffered tile staging: 2 * (16KB values + 16KB indices) = 64KB LDS.
  __shared__ float4 sVal[2][C16_PER_TILE];
  __shared__ int4 sIdx[2][C16_PER_TILE];

  const int tid = (int)threadIdx.x;
  const int stride = (int)gridDim.x;

  const float4* gv = (const float4*)values;
  const int4* gi = (const int4*)indices;
  const float4* gx = (const float4*)x;

  // Cooperative async copy of one tile (16 async-b128 instrs per wave, always
  // exactly 16 so ASYNCcnt bookkeeping stays exact; tail tiles clamp the
  // source address instead of predicating, garbage lands in unused slots).
  auto issue = [&](int tile, int b) {
    size_t base = (size_t)tile * C16_PER_TILE;
    int cnt16 = min(TILE, N - tile * TILE) * 8;
#pragma unroll
    for (int p = 0; p < 8; ++p) {
      int s = p * TILE + tid;
      int g = (s < cnt16) ? s : (cnt16 - 1);
      int d = swz(s);
      async_b128_nt(&gv[base + g], &sVal[b][d]);
      async_b128_nt(&gi[base + g], &sIdx[b][d]);
    }
  };

  int tile = (int)blockIdx.x;
  if (tile < nTiles) issue(tile, 0);

  int buf = 0;
  for (; tile < nTiles; tile += stride) {
    int nxt = tile + stride;
    if (nxt < nTiles) {
      issue(nxt, buf ^ 1);   // prefetch next tile into the other buffer
      wait_async<16>();      // in-order async loads: oldest 16 (this tile) done
    } else {
      wait_async<0>();
    }
    __syncthreads();         // all waves' slices of this tile visible

    int cnt = min(TILE, N - tile * TILE);
    if (tid < cnt) {
      float a0 = 0.f, a1 = 0.f, a2 = 0.f, a3 = 0.f;
      float a4 = 0.f, a5 = 0.f, a6 = 0.f, a7 = 0.f;
#pragma unroll
      for (int j = 0; j < 8; ++j) {
        int s = swz(tid * 8 + j);
        float4 v = sVal[buf][s];
        int4 id = sIdx[buf][s];
        GATHER_FMA(v.x, id.x);
        GATHER_FMA(v.y, id.y);
        GATHER_FMA(v.z, id.z);
        GATHER_FMA(v.w, id.w);
      }
      int n = tile * TILE + tid;
      float bb = bias[n];
      // Write-once output: non-temporal b128 stores, don't pollute L2.
      f32x4* op = (f32x4*)(out + (size_t)n * 8);
      f32x4 o0 = {a0 + bb, a1 + bb, a2 + bb, a3 + bb};
      f32x4 o1 = {a4 + bb, a5 + bb, a6 + bb, a7 + bb};
      __builtin_nontemporal_store(o0, op);
      __builtin_nontemporal_store(o1, op + 1);
    }
    __syncthreads();  // don't let next prefetch overwrite a buffer in use
    buf ^= 1;
  }
}

extern "C" void kernel_launch(void* const* d_in, const int* in_sizes, int n_in,
                              void* d_out, int out_size, void* d_ws,
                              size_t ws_size, hipStream_t stream) {
  // setup_inputs() order: x [N*8 f32], values [N*32 f32], bias [N f32],
  // indices [N*32 i32]
  const float* x = (const float*)d_in[0];
  const float* values = (const float*)d_in[1];
  const float* bias = (const float*)d_in[2];
  const int* indices = (const int*)d_in[3];
  float* out = (float*)d_out;

  int N = in_sizes[2];  // bias length == number of neurons
  int nTiles = (N + TILE - 1) / TILE;
  // ~4 tiles per block -> each block double-buffers across its tiles.
  int blocks = (nTiles + 3) / 4;
  if (blocks < 1) blocks = 1;

  CorticalSheet_spmm_kernel<<<blocks, TILE, 0, stream>>>(
      x, values, bias, indices, out, N, nTiles);
}